// FCOSMultiStrideFilter_83021717832130
// MI455X (gfx1250) — compile-verified
//
#include <hip/hip_runtime.h>
#include <cstdint>

// FCOS multi-stride filter for MI455X (gfx1250).
// Memory-bound NCHW->NHWC transpose + channel max/argmax/threshold.
// TDM (tensor_load_to_lds) DMAs the [C=80 x TILE=128] cls tile
// (row stride = HW) into LDS with 1-dword row padding (stride 129),
// giving bank-conflict-free access for both the per-position channel
// reduction and the transposed score writeback (b128 stores).

#define NB 8       // batch
#define NC 80      // classes
#define TILE 128   // positions per block (= blockDim.x)
#define TILEP 129  // padded LDS row stride (129 mod 64 == 1 -> conflict free)
#define THRESH 0.5f

typedef unsigned int u32;
typedef unsigned long long u64;
typedef __attribute__((ext_vector_type(4))) u32 v4u;
typedef __attribute__((ext_vector_type(8))) int v8i;
typedef __attribute__((ext_vector_type(4))) int v4i;

__global__ __launch_bounds__(TILE) void fcos_filter_level(
    const float* __restrict__ cls, const float* __restrict__ bbox,
    const float* __restrict__ ctr, float* __restrict__ out,
    int HW, int W, long long base)
{
  // single shared array -> static LDS allocation starts at LDS offset 0
  __shared__ float smem[NC * TILEP + TILE];
  float* lds_cls  = smem;               // [NC][TILEP]
  float* lds_mask = smem + NC * TILEP;  // [TILE]

  const int b    = blockIdx.y;
  const int pos0 = blockIdx.x * TILE;
  const int tid  = threadIdx.x;
  const int nvalid = (HW - pos0 < TILE) ? (HW - pos0) : TILE;

  // ---- TDM: DMA cls tile [80 rows x 128 cols], row stride HW, into LDS ----
  if (tid == 0) {
    u64 gaddr = (u64)(size_t)(cls + (size_t)b * NC * HW + pos0);
    u32 td0   = (u32)(HW - pos0);   // X extent for OOB-zero on edge tiles

    // D# group 0: count=1 | lds_addr | global_addr | type=2
    v4u g0 = { 1u,
               0u,                                    // lds_addr = 0
               (u32)(gaddr & 0xFFFFFFFFull),
               (u32)((gaddr >> 32) & 0x01FFFFFFull) | (2u << 30) };

    // D# group 1:
    // w0: data_size=2 (4B) | pad_enable | pad_interval=6 (128 dwords) | pad_amount=0 (1 dword)
    // tensor_dim0 = td0, tensor_dim1 = 80, tile_dim0 = 128, tile_dim1 = 80,
    // tensor_dim0_stride = HW
    v8i g1 = { (int)((2u << 16) | (1u << 20) | (6u << 22)),
               (int)((td0 & 0xFFFFu) << 16),
               (int)((td0 >> 16) | ((u32)NC << 16)),
               (int)(((u32)TILE) << 16),
               (int)NC,
               (int)(u32)HW,
               0, 0 };

    v4i gz4 = { 0, 0, 0, 0 };            // 2D tensor: groups 2/3 unused
    v8i gz8 = { 0, 0, 0, 0, 0, 0, 0, 0 };

    __builtin_amdgcn_tensor_load_to_lds(g0, g1, gz4, gz4, gz8, 0);
    __builtin_amdgcn_s_wait_tensorcnt(0);  // issuing wave waits before barrier
  }
  __syncthreads();

  // ---- per-position channel max / argmax over LDS (lanes vary p: no conflicts) ----
  const bool valid = (tid < nvalid);
  float maxv = -1.0f;
  int maxid = 0;
  if (valid) {
    maxv = lds_cls[tid];
#pragma unroll
    for (int c = 1; c < NC; ++c) {
      float v = lds_cls[c * TILEP + tid];
      maxid = (v > maxv) ? c : maxid;
      maxv  = (v > maxv) ? v : maxv;
    }
  }
  const bool m = valid && (maxv >= THRESH);
  lds_mask[tid] = m ? 1.0f : 0.0f;
  __syncthreads();

  // ---- output section pointers (all float32, concatenated per level) ----
  const long long S = (long long)NB * HW;
  float* out_mv  = out + base;
  float* out_mid = out + base + S;
  float* out_co  = out + base + 2 * S;
  float* out_sc  = out + base + 4 * S;
  float* out_bb  = out + base + (4 + NC) * S;
  float* out_ct  = out + base + (5 + NC) * S;

  const long long rb = (long long)b * HW + pos0;

  if (valid) {
    const int pos = pos0 + tid;
    out_mv [rb + tid] = m ? maxv : 0.0f;
    out_mid[rb + tid] = m ? (float)maxid : -1.0f;

    float2 co = make_float2((float)(pos / W), (float)(pos % W));  // unmasked
    ((float2*)out_co)[rb + tid] = co;

    float cv = ctr[(long long)b * HW + pos];
    out_ct[rb + tid] = m ? cv : 0.0f;

    float4 bbv;
    bbv.x = bbox[((long long)b * 4 + 0) * HW + pos];
    bbv.y = bbox[((long long)b * 4 + 1) * HW + pos];
    bbv.z = bbox[((long long)b * 4 + 2) * HW + pos];
    bbv.w = bbox[((long long)b * 4 + 3) * HW + pos];
    if (!m) bbv = make_float4(0.0f, 0.0f, 0.0f, 0.0f);
    ((float4*)out_bb)[rb + tid] = bbv;
  }

  // ---- transposed score writeback: whole tile is one contiguous span ----
  // out index = rb*C + (p*C + c). n = nvalid*80 is divisible by 4 and the
  // span is 16B-aligned -> float4 stores, fully coalesced (512B/wave).
  // Four consecutive c never cross a row (idx%4==0, 80%4==0), so p is
  // uniform within each float4.
  float4* sc4 = (float4*)(out_sc + rb * NC);
  const int n4 = (nvalid * NC) >> 2;
  for (int i4 = tid; i4 < n4; i4 += TILE) {
    const int idx = i4 << 2;
    const int p = idx / NC;
    const int c = idx - p * NC;
    const float msk = lds_mask[p];
    float4 v;
    if (msk != 0.0f) {
      v.x = lds_cls[(c + 0) * TILEP + p];
      v.y = lds_cls[(c + 1) * TILEP + p];
      v.z = lds_cls[(c + 2) * TILEP + p];
      v.w = lds_cls[(c + 3) * TILEP + p];
    } else {
      v = make_float4(0.0f, 0.0f, 0.0f, 0.0f);
    }
    sc4[i4] = v;
  }
}

extern "C" void kernel_launch(void* const* d_in, const int* in_sizes, int n_in,
                              void* d_out, int out_size, void* d_ws, size_t ws_size,
                              hipStream_t stream) {
  (void)in_sizes; (void)n_in; (void)out_size; (void)d_ws; (void)ws_size;
  static const int HWs[5] = {160 * 160, 80 * 80, 40 * 40, 20 * 20, 10 * 10};
  static const int Ws[5]  = {160, 80, 40, 20, 10};

  long long base = 0;
  for (int i = 0; i < 5; ++i) {
    const float* cls  = (const float*)d_in[3 * i + 0];
    const float* bbox = (const float*)d_in[3 * i + 1];
    const float* ctr  = (const float*)d_in[3 * i + 2];
    const int HW = HWs[i];
    dim3 grid((HW + TILE - 1) / TILE, NB);
    fcos_filter_level<<<grid, TILE, 0, stream>>>(cls, bbox, ctr, (float*)d_out,
                                                 HW, Ws[i], base);
    base += (long long)(4 + NC + 5) * NB * HW;  // 89 outputs per position
  }
}